// ChebNetwork_53987738911396
// MI455X (gfx1250) — compile-verified
//
#include <hip/hip_runtime.h>
#include <hip/hip_bf16.h>

typedef float v2f __attribute__((ext_vector_type(2)));
typedef float v8f __attribute__((ext_vector_type(8)));

#define F_DIM 128   // feature width of x / H1 / H2 (K-dim of every GEMM)

// ---------------------------------------------------------------------------
// Laplacian normalization
// ---------------------------------------------------------------------------
__global__ void cheb_deg_kernel(const int* __restrict__ src, const float* __restrict__ ew,
                                float* __restrict__ deg, int E) {
    int e = blockIdx.x * blockDim.x + threadIdx.x;
    if (e < E) atomicAdd(&deg[src[e]], ew[e]);
}

__global__ void cheb_node_norm_kernel(const float* __restrict__ deg, float* __restrict__ dis,
                                      float* __restrict__ diag, int N) {
    int n = blockIdx.x * blockDim.x + threadIdx.x;
    if (n < N) {
        float d = deg[n];
        bool pos = d > 0.0f;
        dis[n]  = pos ? rsqrtf(fmaxf(d, 1e-30f)) : 0.0f;
        // (2/lambda_max) * 1[deg>0] - 1, with lambda_max = 2.0
        diag[n] = (pos ? 1.0f : 0.0f) - 1.0f;
    }
}

__global__ void cheb_edge_norm_kernel(const int* __restrict__ src, const int* __restrict__ dst,
                                      const float* __restrict__ ew, const float* __restrict__ dis,
                                      float* __restrict__ lapw, int E) {
    int e = blockIdx.x * blockDim.x + threadIdx.x;
    if (e < E) lapw[e] = -dis[src[e]] * ew[e] * dis[dst[e]];  // *(2/lambda_max)==1
}

// ---------------------------------------------------------------------------
// Sparse propagation: out = alpha*diag[n]*tin + beta*tsub   (full overwrite)
// then edge scatter: out[dst] += scale*lapw[e]*tin[src]
// ---------------------------------------------------------------------------
__global__ void cheb_prop_init_kernel(float* __restrict__ out, const float* __restrict__ tin,
                                      const float* __restrict__ tsub, const float* __restrict__ diag,
                                      float alpha, float beta, int total) {
    int i = blockIdx.x * blockDim.x + threadIdx.x;
    if (i < total) {
        int n = i >> 7;  // /F_DIM
        out[i] = alpha * diag[n] * tin[i] + beta * tsub[i];
    }
}

// one wave32 per edge: lane l covers features [4l .. 4l+3] (32*4 == 128)
__global__ void cheb_prop_edge_kernel(float* __restrict__ out, const float* __restrict__ tin,
                                      const int* __restrict__ src, const int* __restrict__ dst,
                                      const float* __restrict__ lapw, float scale, int E) {
    int e    = (blockIdx.x * blockDim.x + threadIdx.x) >> 5;  // wave-uniform
    int lane = threadIdx.x & 31;
    if (e >= E) return;
    int s = src[e], d = dst[e];
    float w = scale * lapw[e];
    float4 v = ((const float4*)(tin + (size_t)s * F_DIM))[lane];
    float* drow = out + (size_t)d * F_DIM + lane * 4;
    atomicAdd(drow + 0, w * v.x);
    atomicAdd(drow + 1, w * v.y);
    atomicAdd(drow + 2, w * v.z);
    atomicAdd(drow + 3, w * v.w);
}

// ---------------------------------------------------------------------------
// Fused Chebyshev GEMM: out = sigmoid(T0@W[0] + T1@W[1] + T2@W[2] + b)
//
// Grid: (colTiles, ceil(rowTiles/8)). Block = 256 threads = 8 waves.
// All 8 waves share one 16-column tile; the block stages the needed
// W slice (3 x 128 x 16 f32 = 24 KB) into LDS once, then each wave
// computes one 16x16 output tile with V_WMMA_F32_16X16X4_F32.
// HOUT is a template constant so all global/LDS addresses use
// immediate offsets (no 64-bit multiplies in the inner loop).
//
// f32 WMMA operand layout (ISA 7.12.2):
//   A 16x4 : lanes 0-15 -> M=lane, VGPR{0,1}=K{0,1}; lanes 16-31 -> K{2,3}
//   B 4x16 : mirrored (lanes 0-15 hold N=lane, K{0,1}; lanes 16-31 K{2,3})
//   C/D    : VGPR v -> M = v + 8*(lane>=16), N = lane&15
// ---------------------------------------------------------------------------
template <int HOUT>
__global__ __launch_bounds__(256)
void cheb_gemm_kernel(const float* __restrict__ T0, const float* __restrict__ T1,
                      const float* __restrict__ T2, const float* __restrict__ W,
                      const float* __restrict__ bias, float* __restrict__ out,
                      int rowTiles) {
    __shared__ float sW[3 * F_DIM * 16];   // [t*128+k][c], 24 KB

    const int tid     = threadIdx.x;
    const int lane    = tid & 31;
    const int waveId  = tid >> 5;
    const int laneLo  = lane & 15;
    const int hiSel   = lane >> 4;         // 0 or 1
    const int colBase = blockIdx.x << 4;
    const int rowTile = blockIdx.y * 8 + waveId;

    // ---- cooperative LDS stage of W[:, :, colBase:colBase+16] ----
    // global: W[(t*128+k)*HOUT + colBase + c]  ->  sW[(t*128+k)*16 + c]
#pragma unroll
    for (int idx = 0; idx < 3 * F_DIM * 16; idx += 256) {
        const int i    = idx + tid;
        const int kLin = i >> 4;           // 0..383  (= t*128 + k)
        const int c    = i & 15;
        sW[i] = W[kLin * HOUT + colBase + c];
    }
    __syncthreads();

    if (rowTile >= rowTiles) return;       // wave-uniform (after barrier)
    const int rowBase = rowTile << 4;

    v8f c = {};
    const float* Ts[3] = {T0, T1, T2};
#pragma unroll
    for (int t = 0; t < 3; ++t) {
        // per-lane bases; inner-loop offsets are compile-time immediates
        const float* A  = Ts[t] + (size_t)(rowBase + laneLo) * F_DIM + (hiSel << 1);
        const float* Bl = sW + t * (F_DIM * 16) + (hiSel << 5) + laneLo;
#pragma unroll 8
        for (int k0 = 0; k0 < F_DIM; k0 += 4) {
            v2f a, b;
            a.x = A[k0];
            a.y = A[k0 + 1];
            b.x = Bl[k0 * 16];
            b.y = Bl[k0 * 16 + 16];
            c = __builtin_amdgcn_wmma_f32_16x16x4_f32(false, a, false, b,
                                                      (short)0, c, false, false);
        }
    }

    const int col = colBase + laneLo;
    const float bv = bias[col];
#pragma unroll
    for (int v = 0; v < 8; ++v) {
        const int row = rowBase + v + hiSel * 8;
        const float val = c[v] + bv;
        out[(size_t)row * HOUT + col] = 1.0f / (1.0f + __expf(-val));
    }
}

// ---------------------------------------------------------------------------
// Host-side orchestration (graph-capture safe: kernels + hipMemsetAsync only)
// ---------------------------------------------------------------------------
extern "C" void kernel_launch(void* const* d_in, const int* in_sizes, int n_in,
                              void* d_out, int out_size, void* d_ws, size_t ws_size,
                              hipStream_t stream) {
    const float* x  = (const float*)d_in[0];
    const int*   ei = (const int*)d_in[1];   // [2, E] int32
    const float* ew = (const float*)d_in[2];
    const float* W0 = (const float*)d_in[3];
    const float* b0 = (const float*)d_in[4];
    const float* W1 = (const float*)d_in[5];
    const float* b1 = (const float*)d_in[6];
    const float* W2 = (const float*)d_in[7];
    const float* b2 = (const float*)d_in[8];

    const int N = in_sizes[0] / F_DIM;
    const int E = in_sizes[2];
    const int* src = ei;       // edge_index[0] (gather rows, degree rows)
    const int* dst = ei + E;   // edge_index[1] (scatter rows)
    float* out = (float*)d_out;

    // workspace layout (floats): T1, T2, H1, H2 (N*128 each), lapw (E),
    // deg, dis, diag (N each)  ~= 219 MB total (L2 is 192 MB; hot set =
    // the 3 T buffers + lapw ~= 166 MB stays L2-resident during scatter)
    float* T1   = (float*)d_ws;
    float* T2   = T1   + (size_t)N * F_DIM;
    float* H1m  = T2   + (size_t)N * F_DIM;
    float* H2m  = H1m  + (size_t)N * F_DIM;
    float* lapw = H2m  + (size_t)N * F_DIM;
    float* deg  = lapw + (size_t)E;
    float* dis  = deg  + (size_t)N;
    float* diag = dis  + (size_t)N;

    const int gE = (E + 255) / 256;
    const int gN = (N + 255) / 256;

    // ---- Laplacian normalization (shared by all 3 layers) ----
    hipMemsetAsync(deg, 0, (size_t)N * sizeof(float), stream);
    cheb_deg_kernel<<<gE, 256, 0, stream>>>(src, ew, deg, E);
    cheb_node_norm_kernel<<<gN, 256, 0, stream>>>(deg, dis, diag, N);
    cheb_edge_norm_kernel<<<gE, 256, 0, stream>>>(src, dst, ew, dis, lapw, E);

    const int total    = N * F_DIM;
    const int gInit    = (total + 255) / 256;
    const int gEdge    = (E + 7) / 8;        // 8 waves (edges) per block
    const int rowTiles = N >> 4;
    const dim3 blk(256);

    auto props = [&](const float* Tin) {
        // T1 = L_hat @ Tin
        cheb_prop_init_kernel<<<gInit, blk, 0, stream>>>(T1, Tin, Tin, diag, 1.0f, 0.0f, total);
        cheb_prop_edge_kernel<<<gEdge, blk, 0, stream>>>(T1, Tin, src, dst, lapw, 1.0f, E);
        // T2 = 2 * (L_hat @ T1) - Tin
        cheb_prop_init_kernel<<<gInit, blk, 0, stream>>>(T2, T1, Tin, diag, 2.0f, -1.0f, total);
        cheb_prop_edge_kernel<<<gEdge, blk, 0, stream>>>(T2, T1, src, dst, lapw, 2.0f, E);
    };

    const dim3 g128(128 / 16, (rowTiles + 7) / 8);
    const dim3 g64(64 / 16, (rowTiles + 7) / 8);

    // layer 1: H1 = sigmoid(cheb(x) @ W0 + b0)
    props(x);
    cheb_gemm_kernel<128><<<g128, blk, 0, stream>>>(x, T1, T2, W0, b0, H1m, rowTiles);
    // layer 2: H2 = sigmoid(cheb(H1) @ W1 + b1)
    props(H1m);
    cheb_gemm_kernel<128><<<g128, blk, 0, stream>>>(H1m, T1, T2, W1, b1, H2m, rowTiles);
    // layer 3: out = sigmoid(cheb(H2) @ W2 + b2)
    props(H2m);
    cheb_gemm_kernel<64><<<g64, blk, 0, stream>>>(H2m, T1, T2, W2, b2, out, rowTiles);
}